// RnCLoss_v2_79147657331272
// MI455X (gfx1250) — compile-verified
//
#include <hip/hip_runtime.h>
#include <math.h>

// ---------------------------------------------------------------------------
// RnC loss for MI455X (gfx1250, wave32).
// Heavy op: Gram matrix F*F^T (512x512x256 fp32) done with
// V_WMMA_F32_16X16X4_F32, one wave per 16x16 output tile.
// ---------------------------------------------------------------------------

#define N 512
#define D 256
#define TEMPERATURE 2.0f
#define RAD2DEG 57.295779513082320876f

typedef __attribute__((ext_vector_type(2))) float v2f;
typedef __attribute__((ext_vector_type(8))) float v8f;

// ---------------------------------------------------------------------------
// Kernel 1: pose preprocessing (rot6d -> normalized 3x3 rot + t) and zero out.
// ---------------------------------------------------------------------------
__global__ void prep_kernel(const float* __restrict__ labels,
                            float* __restrict__ rot,
                            float* __restrict__ tvec,
                            float* __restrict__ out) {
  int i = blockIdx.x * blockDim.x + threadIdx.x;
  if (i == 0) out[0] = 0.0f;  // reset accumulator every launch (graph replay safe)
  if (i >= N) return;

  const float* lab = labels + i * 9;
  float a1x = lab[0], a1y = lab[1], a1z = lab[2];
  float a2x = lab[3], a2y = lab[4], a2z = lab[5];

  float n1 = sqrtf(a1x * a1x + a1y * a1y + a1z * a1z);
  float b1x = a1x / n1, b1y = a1y / n1, b1z = a1z / n1;

  float dp = b1x * a2x + b1y * a2y + b1z * a2z;
  float px = a2x - dp * b1x, py = a2y - dp * b1y, pz = a2z - dp * b1z;
  float n2 = sqrtf(px * px + py * py + pz * pz);
  float b2x = px / n2, b2y = py / n2, b2z = pz / n2;

  float b3x = b1y * b2z - b1z * b2y;
  float b3y = b1z * b2x - b1x * b2z;
  float b3z = b1x * b2y - b1y * b2x;

  // det of rows [b1; b2; b3]
  float det = b1x * (b2y * b3z - b2z * b3y)
            - b1y * (b2x * b3z - b2z * b3x)
            + b1z * (b2x * b3y - b2y * b3x);
  float inv = 1.0f / powf(det, 1.0f / 3.0f);

  float* R = rot + i * 9;
  R[0] = b1x * inv; R[1] = b1y * inv; R[2] = b1z * inv;
  R[3] = b2x * inv; R[4] = b2y * inv; R[5] = b2z * inv;
  R[6] = b3x * inv; R[7] = b3y * inv; R[8] = b3z * inv;

  float* t = tvec + i * 3;
  t[0] = lab[6]; t[1] = lab[7]; t[2] = lab[8];
}

// ---------------------------------------------------------------------------
// Kernel 2: per-row squared norms of features. One wave (32 lanes) per row.
// ---------------------------------------------------------------------------
__global__ void norm_kernel(const float* __restrict__ F,
                            float* __restrict__ nrm) {
  int wave = threadIdx.x >> 5;              // 8 waves / block
  int lane = threadIdx.x & 31;
  int row = blockIdx.x * 8 + wave;
  const float* fr = F + row * D;
  float s = 0.0f;
#pragma unroll
  for (int q = 0; q < D / 32; ++q) {
    float v = fr[lane + 32 * q];
    s += v * v;
  }
#pragma unroll
  for (int m = 16; m > 0; m >>= 1) s += __shfl_xor(s, m, 32);
  if (lane == 0) nrm[row] = s;
}

// ---------------------------------------------------------------------------
// Kernel 3: Gram via WMMA f32 16x16x4, then logits = -safe_norm(fi-fj)/T.
// One wave per 16x16 tile; grid = (N/16, N/16).
//
// A fragment (16x4, MxK):  lane l: m=l&15, g=l>>4; a[v] = F[ti*16+m][k+2g+v]
// B fragment (4x16, KxN):  lane l: n=l&15, g=l>>4; b[v] = F[tj*16+n][k+2g+v]
//   (identical addressing because B = F^T for the Gram product)
// C/D (16x16 f32): lane l: n=l&15, g=l>>4; VGPR r holds row m = r + 8g.
// ---------------------------------------------------------------------------
__global__ void gram_logits_kernel(const float* __restrict__ F,
                                   const float* __restrict__ nrm,
                                   float* __restrict__ logits) {
  int ti = blockIdx.y, tj = blockIdx.x;
  int lane = threadIdx.x;
  int m = lane & 15;
  int g = lane >> 4;

  const float* Arow = F + (ti * 16 + m) * D + 2 * g;
  const float* Brow = F + (tj * 16 + m) * D + 2 * g;

  v8f c0 = {};
  v8f c1 = {};
#pragma unroll 4
  for (int k = 0; k < D; k += 8) {
    v2f a0, b0, a1, b1;
    a0.x = Arow[k];     a0.y = Arow[k + 1];
    b0.x = Brow[k];     b0.y = Brow[k + 1];
    a1.x = Arow[k + 4]; a1.y = Arow[k + 5];
    b1.x = Brow[k + 4]; b1.y = Brow[k + 5];
    c0 = __builtin_amdgcn_wmma_f32_16x16x4_f32(false, a0, false, b0,
                                               (short)0, c0, false, false);
    c1 = __builtin_amdgcn_wmma_f32_16x16x4_f32(false, a1, false, b1,
                                               (short)0, c1, false, false);
  }

  int j = tj * 16 + m;
  float nj = nrm[j];
#pragma unroll
  for (int r = 0; r < 8; ++r) {
    int i = ti * 16 + r + 8 * g;
    float dot = c0[r] + c1[r];
    float d2 = nrm[i] + nj - 2.0f * dot;
    float dist = (d2 > 0.0f) ? sqrtf(d2) : 0.0f;           // safe_norm
    logits[i * N + j] = (i == j) ? 0.0f : (-dist / TEMPERATURE);
  }
}

// ---------------------------------------------------------------------------
// Kernel 4: pairwise pose errors: theta (deg) and shift (x100).
// ---------------------------------------------------------------------------
__global__ void pair_kernel(const float* __restrict__ rot,
                            const float* __restrict__ tvec,
                            float* __restrict__ TD,
                            float* __restrict__ SD) {
  int tid = blockIdx.x * blockDim.x + threadIdx.x;
  int i = tid >> 9;          // /N
  int j = tid & (N - 1);     // %N
  const float* Ri = rot + i * 9;
  const float* Rj = rot + j * 9;
  float tr = 0.0f;
#pragma unroll
  for (int a = 0; a < 9; ++a) tr += Ri[a] * Rj[a];
  float ct = 0.5f * (tr - 1.0f);
  ct = fminf(1.0f, fmaxf(-1.0f, ct));
  TD[tid] = acosf(ct) * RAD2DEG;

  const float* ti = tvec + i * 3;
  const float* tj = tvec + j * 3;
  float dx = ti[0] - tj[0], dy = ti[1] - tj[1], dz = ti[2] - tj[2];
  float d2 = dx * dx + dy * dy + dz * dz;
  SD[tid] = (d2 > 0.0f) ? sqrtf(d2) * 100.0f : 0.0f;       // safe_norm * 100
}

// ---------------------------------------------------------------------------
// Kernel 5: per-row denominators + loss accumulation. One block per row i.
// E[i]=0 implements the diagonal exclusion done by take_along_axis in the ref.
// Row-max subtraction skipped: diagonal logit == 0 and all logits <= 0.
// ---------------------------------------------------------------------------
__global__ void loss_kernel(const float* __restrict__ logits,
                            const float* __restrict__ TD,
                            const float* __restrict__ SD,
                            float* __restrict__ out) {
  __shared__ float sE[N];
  __shared__ float sT[N];
  __shared__ float sS[N];
  __shared__ float red[256];

  int i = blockIdx.x;
  int tid = threadIdx.x;

  for (int k = tid; k < N; k += 256) {
    float l = logits[i * N + k];
    sE[k] = (k == i) ? 0.0f : expf(l);
    sT[k] = TD[i * N + k];
    sS[k] = SD[i * N + k];
  }
  __syncthreads();

  float acc = 0.0f;
  for (int j = tid; j < N; j += 256) {
    if (j == i) continue;
    float tdj = sT[j];
    float sdj = sS[j];
    float dt = 0.0f, ds = 0.0f;
    for (int k = 0; k < N; ++k) {
      float e = sE[k];
      dt += (sT[k] >= tdj) ? e : 0.0f;
      ds += (sS[k] >= sdj) ? e : 0.0f;
    }
    acc += 2.0f * logits[i * N + j] - logf(dt) - logf(ds);
  }

  red[tid] = acc;
  __syncthreads();
#pragma unroll
  for (int s = 128; s > 0; s >>= 1) {
    if (tid < s) red[tid] += red[tid + s];
    __syncthreads();
  }
  if (tid == 0) {
    // pos_log_probs = acc/2 ; loss = -sum / (n*(n-1))
    atomicAdd(out, -red[0] / (2.0f * (float)N * (float)(N - 1)));
  }
}

// ---------------------------------------------------------------------------
// Launch
// ---------------------------------------------------------------------------
extern "C" void kernel_launch(void* const* d_in, const int* in_sizes, int n_in,
                              void* d_out, int out_size, void* d_ws, size_t ws_size,
                              hipStream_t stream) {
  const float* features = (const float*)d_in[0];   // (512, 256) f32
  const float* labels   = (const float*)d_in[1];   // (512, 9)   f32
  float* out = (float*)d_out;                      // scalar f32

  // workspace layout (floats): rot[N*9] | t[N*3] | nrm[N] | logits[N*N] | TD[N*N] | SD[N*N]
  float* ws     = (float*)d_ws;
  float* rot    = ws;
  float* tvec   = ws + N * 9;
  float* nrm    = ws + N * 12;
  float* logits = ws + N * 13;
  float* TD     = logits + (size_t)N * N;
  float* SD     = TD + (size_t)N * N;

  prep_kernel<<<dim3((N + 63) / 64), dim3(64), 0, stream>>>(labels, rot, tvec, out);
  norm_kernel<<<dim3(N / 8), dim3(256), 0, stream>>>(features, nrm);
  gram_logits_kernel<<<dim3(N / 16, N / 16), dim3(32), 0, stream>>>(features, nrm, logits);
  pair_kernel<<<dim3((N * N) / 256), dim3(256), 0, stream>>>(rot, tvec, TD, SD);
  loss_kernel<<<dim3(N), dim3(256), 0, stream>>>(logits, TD, SD, out);
}